// LSTMLayer_57681410786154
// MI455X (gfx1250) — compile-verified
//
#include <hip/hip_runtime.h>
#include <hip/hip_bf16.h>
#include <math.h>

// LSTM: SEQ=64, B=40 (pad 48), IN=16384, H=1024, 4H=4096. fp32 throughout.
#define SEQ   64
#define BATCH 40
#define BPAD  48
#define INDIM 16384
#define HDIM  1024
#define GDIM  4096   // 4*H

typedef float v2f __attribute__((ext_vector_type(2)));
typedef float v8f __attribute__((ext_vector_type(8)));

__device__ __forceinline__ v8f wmma_f32_k4(v2f a, v2f b, v8f c) {
    // V_WMMA_F32_16X16X4_F32 : D = A(16x4) x B(4x16) + C(16x16), fp32
    return __builtin_amdgcn_wmma_f32_16x16x4_f32(
        /*neg_a=*/false, a, /*neg_b=*/false, b,
        /*c_mod=*/(short)0, c, /*reuse_a=*/false, /*reuse_b=*/false);
}

__device__ __forceinline__ float sigmoidf_dev(float x) {
    return 1.0f / (1.0f + __expf(-x));
}

// ---------------------------------------------------------------------------
// Zero-init padded h (both buffers) and c state. n = total floats.
// ---------------------------------------------------------------------------
__global__ void lstm_zero_state(float* __restrict__ p, int n) {
    int i = blockIdx.x * blockDim.x + threadIdx.x;
    if (i < n) p[i] = 0.0f;
}

// ---------------------------------------------------------------------------
// Phase 1: xw[64,4096] = features[64,16384] @ w_ih[4096,16384]^T + (b_ih+b_hh)
// Grid: 256 blocks (one 16-wide N strip each), 256 threads = 8 waves.
// wave = mtile (0..3) + 4*ksplit (0..1). Split-K reduced through LDS.
// w_ih (256 MB, read-once) is loaded with NT temporal hint so the one-shot
// weight stream does not evict the L2-resident features (4 MB, re-read by
// all 256 blocks), w_hh, or xw working set.
// ---------------------------------------------------------------------------
__global__ void __launch_bounds__(256)
lstm_xw_gemm(const float* __restrict__ features,
             const float* __restrict__ w_ih,
             const float* __restrict__ b_ih,
             const float* __restrict__ b_hh,
             float* __restrict__ xw) {
    __shared__ float red[4][16][16];            // 4 KB split-K exchange

    const int tid    = threadIdx.x;
    const int wave   = tid >> 5;
    const int lane   = tid & 31;
    const int l15    = lane & 15;
    const int hi     = lane >> 4;               // half-wave select
    const int mtile  = wave & 3;                // rows 16*mtile .. +15
    const int ksplit = wave >> 2;               // 0 or 1 (8192 K each)
    const int n0     = blockIdx.x * 16;

    const int m = mtile * 16 + l15;             // A row held by this lane
    const int n = n0 + l15;                     // B column held by this lane

    const float* __restrict__ aptr =
        features + (size_t)m * INDIM + (size_t)ksplit * 8192 + 2 * hi;
    const float* __restrict__ bptr =
        w_ih     + (size_t)n * INDIM + (size_t)ksplit * 8192 + 2 * hi;

    v8f acc0 = {};
    v8f acc1 = {};
    for (int k = 0; k < 8192; k += 8) {
        v2f a0 = *(const v2f*)(aptr + k);
        v2f b0 = __builtin_nontemporal_load((const v2f*)(bptr + k));
        acc0 = wmma_f32_k4(a0, b0, acc0);
        v2f a1 = *(const v2f*)(aptr + k + 4);
        v2f b1 = __builtin_nontemporal_load((const v2f*)(bptr + k + 4));
        acc1 = wmma_f32_k4(a1, b1, acc1);
    }
    v8f acc = acc0 + acc1;

    if (ksplit == 1) {
        #pragma unroll
        for (int v = 0; v < 8; ++v)
            red[mtile][v + 8 * hi][l15] = acc[v];
    }
    __syncthreads();
    if (ksplit == 0) {
        const float bias = b_ih[n] + b_hh[n];
        #pragma unroll
        for (int v = 0; v < 8; ++v) {
            const int mm = mtile * 16 + v + 8 * hi;
            xw[(size_t)mm * GDIM + n0 + l15] =
                acc[v] + red[mtile][v + 8 * hi][l15] + bias;
        }
    }
}

// ---------------------------------------------------------------------------
// Phase 2: one recurrence step.
// gates[48,4096] = xw_t (bcast over rows) + h_in[48,1024] @ w_hh[4096,1024]^T
// then i,f,g,o -> c,h update fused via LDS.
// Grid: 64 blocks (16 hidden cols each), 384 threads = 12 waves:
//   wave = gate*3 + mtile   (gate 0..3 = i,f,g,o ; mtile 0..2 = rows 0..47)
// w_hh (16 MB) stays L2-resident across all 64 steps (192 MB L2), so only
// the first step pays HBM latency for it.
// ---------------------------------------------------------------------------
__global__ void __launch_bounds__(384)
lstm_step(const float* __restrict__ w_hh,
          const float* __restrict__ xw_t,      // xw + t*4096
          const float* __restrict__ h_in,      // [48,1024] padded state
          float* __restrict__ h_next,          // [48,1024] padded state
          float* __restrict__ c_st,            // [48,1024] padded cell state
          float* __restrict__ h_glb) {         // d_out + t*40*1024
    __shared__ float gt[4][3][16][16];          // 12 KB gate tiles

    const int tid   = threadIdx.x;
    const int wave  = tid >> 5;
    const int lane  = tid & 31;
    const int l15   = lane & 15;
    const int hi    = lane >> 4;
    const int mtile = wave % 3;
    const int gate  = wave / 3;
    const int j0    = blockIdx.x * 16;          // hidden column strip

    const int n = gate * HDIM + j0 + l15;       // gate column in [0,4096)
    const int m = mtile * 16 + l15;             // state row in [0,48)

    const float* __restrict__ aptr = h_in + (size_t)m * HDIM + 2 * hi;
    const float* __restrict__ bptr = w_hh + (size_t)n * HDIM + 2 * hi;

    v8f acc0 = {};
    v8f acc1 = {};
    for (int k = 0; k < HDIM; k += 8) {
        v2f a0 = *(const v2f*)(aptr + k);
        v2f b0 = *(const v2f*)(bptr + k);
        acc0 = wmma_f32_k4(a0, b0, acc0);
        v2f a1 = *(const v2f*)(aptr + k + 4);
        v2f b1 = *(const v2f*)(bptr + k + 4);
        acc1 = wmma_f32_k4(a1, b1, acc1);
    }
    v8f acc = acc0 + acc1;

    const float xb = xw_t[n];                   // hoisted bias+input projection
    #pragma unroll
    for (int v = 0; v < 8; ++v)
        gt[gate][mtile][v + 8 * hi][l15] = acc[v] + xb;
    __syncthreads();

    // Fused LSTM cell update for the 40 valid rows x 16 cols of this strip.
    for (int e = tid; e < BATCH * 16; e += 384) {
        const int row = e >> 4;                 // 0..39
        const int col = e & 15;
        const int rt  = row >> 4;               // row tile
        const int rl  = row & 15;
        const float gi = gt[0][rt][rl][col];
        const float gf = gt[1][rt][rl][col];
        const float gg = gt[2][rt][rl][col];
        const float go = gt[3][rt][rl][col];

        const float i = sigmoidf_dev(gi);
        const float f = sigmoidf_dev(gf);
        const float g = tanhf(gg);
        const float o = sigmoidf_dev(go);

        const size_t sidx = (size_t)row * HDIM + j0 + col;
        const float c = f * c_st[sidx] + i * g;
        const float h = o * tanhf(c);
        c_st[sidx]   = c;
        h_next[sidx] = h;
        h_glb[sidx]  = h;                       // d_out is [SEQ,40,1024]
    }
}

// ---------------------------------------------------------------------------
// Launch: zero state -> big GEMM -> 64 sequential recurrence steps.
// ws layout (floats): xw[64*4096] | hA[48*1024] | hB[48*1024] | c[48*1024]
// ---------------------------------------------------------------------------
extern "C" void kernel_launch(void* const* d_in, const int* in_sizes, int n_in,
                              void* d_out, int out_size, void* d_ws, size_t ws_size,
                              hipStream_t stream) {
    const float* features = (const float*)d_in[0];
    const float* w_ih     = (const float*)d_in[1];
    const float* w_hh     = (const float*)d_in[2];
    const float* b_ih     = (const float*)d_in[3];
    const float* b_hh     = (const float*)d_in[4];
    float* out = (float*)d_out;
    float* ws  = (float*)d_ws;

    float* xw  = ws;                               // 262144 floats
    float* hA  = xw + (size_t)SEQ * GDIM;          //  49152 floats
    float* hB  = hA + (size_t)BPAD * HDIM;         //  49152 floats
    float* cst = hB + (size_t)BPAD * HDIM;         //  49152 floats

    // Zero hA, hB, c (contiguous 3*49152 floats). Rows 40..47 stay zero so
    // padded WMMA A-reads are well defined.
    const int zn = 3 * BPAD * HDIM;
    lstm_zero_state<<<(zn + 255) / 256, 256, 0, stream>>>(hA, zn);

    // Phase 1: input projection (HBM-bound; w_ih read exactly once, NT hint).
    lstm_xw_gemm<<<GDIM / 16, 256, 0, stream>>>(features, w_ih, b_ih, b_hh, xw);

    // Phase 2: 64 sequential steps; h double-buffered, c owner-written.
    for (int t = 0; t < SEQ; ++t) {
        float* hin  = (t & 1) ? hB : hA;
        float* hout = (t & 1) ? hA : hB;
        lstm_step<<<HDIM / 16, 384, 0, stream>>>(
            w_hh, xw + (size_t)t * GDIM, hin, hout, cst,
            out + (size_t)t * BATCH * HDIM);
    }
}